// MRCAGCFN_36361193128543
// MI455X (gfx1250) — compile-verified
//
#include <hip/hip_runtime.h>
#include <math.h>

// ---------------------------------------------------------------------------
// Problem constants (from reference): S=200, L=9, N=81, HID=128, NCLS=16, B=32
// ---------------------------------------------------------------------------
#define SDIM   200
#define LDIM   9
#define NNODE  81
#define NPAD   84                // NNODE rounded up to multiple of 4 (K padding)
#define HIDC   128
#define BATCH  32
#define NCLSO  16
#define ROWS   (BATCH * NNODE)   // 2592
#define EPSV   1e-5f
#define CH_B   4                 // images per deform-conv im2col chunk

typedef __attribute__((ext_vector_type(2))) float v2f;
typedef __attribute__((ext_vector_type(8))) float v8f;

__device__ __forceinline__ float mishf(float x) {
    // softplus(x) = log1p(exp(x)), stable form; mish = x * tanh(softplus(x))
    float sp = fmaxf(x, 0.0f) + log1pf(expf(-fabsf(x)));
    return x * tanhf(sp);
}

// ---------------------------------------------------------------------------
// fp32 WMMA GEMM:  C[m,n] = sum_k A[m,k] * B(k,n)  (+ bias[n])
//   A: M x K row-major.  B_NXK=true: B is N x K row-major (weights);
//   B_NXK=false: B is K x N row-major (activations).
//   Requirements (guaranteed by caller): K % 4 == 0 (ragged K pre-padded with
//   zeros); N % 32 == 0 or harmless.  Load indices are CLAMPED (never guarded)
//   so the inner loop is branchless: clamped duplicate rows/cols only feed C
//   elements that the store guard discards.
//   Each wave computes a 32x32 C tile: 2 A frags x 2 B frags -> 4 WMMA per
//   K-slab (V_WMMA_F32_16X16X4_F32), A/B frags loaded as 8B-aligned float2.
//   blockDim = (32,4): 4 waves tile 128 rows of M.  Batched via gridDim.z.
// ---------------------------------------------------------------------------
template <bool B_NXK>
__global__ __launch_bounds__(128)
void wmma_gemm_f32(const float* __restrict__ A,
                   const float* __restrict__ Bm,
                   float* __restrict__ C,
                   const float* __restrict__ bias,
                   int M, int N, int K,
                   long sA, long sB, long sC) {
    const float* Ab = A  + (long)blockIdx.z * sA;
    const float* Bb = Bm + (long)blockIdx.z * sB;
    float*       Cb = C  + (long)blockIdx.z * sC;

    const int lane  = threadIdx.x;                 // 0..31
    const int l15   = lane & 15;
    const int khalf = (lane >> 4) * 2;             // 0 or 2 (K sub-slab select)
    const int m0    = (blockIdx.x * 4 + threadIdx.y) * 32;
    const int n0    = blockIdx.y * 32;

    // Clamped (always in-bounds) load rows/cols -> branchless loop.
    const int mA0 = min(m0 + l15,      M - 1);
    const int mA1 = min(m0 + 16 + l15, M - 1);
    const int nB0 = min(n0 + l15,      N - 1);
    const int nB1 = min(n0 + 16 + l15, N - 1);

    const float* a0p = Ab + (long)mA0 * K + khalf;
    const float* a1p = Ab + (long)mA1 * K + khalf;

    v8f acc00 = {}, acc01 = {}, acc10 = {}, acc11 = {};

    if (B_NXK) {
        const float* b0p = Bb + (long)nB0 * K + khalf;
        const float* b1p = Bb + (long)nB1 * K + khalf;
#pragma unroll 2
        for (int k = 0; k < K; k += 4) {
            const v2f av0 = *(const v2f*)(a0p + k);
            const v2f av1 = *(const v2f*)(a1p + k);
            const v2f bv0 = *(const v2f*)(b0p + k);
            const v2f bv1 = *(const v2f*)(b1p + k);
            acc00 = __builtin_amdgcn_wmma_f32_16x16x4_f32(false, av0, false, bv0,
                                                          (short)0, acc00, false, false);
            acc01 = __builtin_amdgcn_wmma_f32_16x16x4_f32(false, av0, false, bv1,
                                                          (short)0, acc01, false, false);
            acc10 = __builtin_amdgcn_wmma_f32_16x16x4_f32(false, av1, false, bv0,
                                                          (short)0, acc10, false, false);
            acc11 = __builtin_amdgcn_wmma_f32_16x16x4_f32(false, av1, false, bv1,
                                                          (short)0, acc11, false, false);
        }
    } else {
#pragma unroll 2
        for (int k = 0; k < K; k += 4) {
            const v2f av0 = *(const v2f*)(a0p + k);
            const v2f av1 = *(const v2f*)(a1p + k);
            const int k0 = k + khalf;
            v2f bv0, bv1;
            bv0.x = Bb[(long)k0 * N + nB0];
            bv0.y = Bb[(long)(k0 + 1) * N + nB0];
            bv1.x = Bb[(long)k0 * N + nB1];
            bv1.y = Bb[(long)(k0 + 1) * N + nB1];
            acc00 = __builtin_amdgcn_wmma_f32_16x16x4_f32(false, av0, false, bv0,
                                                          (short)0, acc00, false, false);
            acc01 = __builtin_amdgcn_wmma_f32_16x16x4_f32(false, av0, false, bv1,
                                                          (short)0, acc01, false, false);
            acc10 = __builtin_amdgcn_wmma_f32_16x16x4_f32(false, av1, false, bv0,
                                                          (short)0, acc10, false, false);
            acc11 = __builtin_amdgcn_wmma_f32_16x16x4_f32(false, av1, false, bv1,
                                                          (short)0, acc11, false, false);
        }
    }

    // C/D layout: vgpr r, lanes 0-15 -> row base+r; lanes 16-31 -> row base+8+r.
    const int half8 = (lane >> 4) * 8;
    const int nc0 = n0 + l15;
    const int nc1 = n0 + 16 + l15;
    const float bs0 = (bias && nc0 < N) ? bias[nc0] : 0.0f;
    const float bs1 = (bias && nc1 < N) ? bias[nc1] : 0.0f;
#pragma unroll
    for (int r = 0; r < 8; ++r) {
        const int mr0 = m0 + half8 + r;
        const int mr1 = m0 + 16 + half8 + r;
        if (mr0 < M) {
            if (nc0 < N) Cb[(long)mr0 * N + nc0] = acc00[r] + bs0;
            if (nc1 < N) Cb[(long)mr0 * N + nc1] = acc01[r] + bs1;
        }
        if (mr1 < M) {
            if (nc0 < N) Cb[(long)mr1 * N + nc0] = acc10[r] + bs0;
            if (nc1 < N) Cb[(long)mr1 * N + nc1] = acc11[r] + bs1;
        }
    }
}

// ---------------------------------------------------------------------------
// K-padding for the dist GEMM (K = 81 -> 84).
// DP (B, 81, 84): tail columns zeroed.  XGP (B, 84, 128): tail rows zeroed.
// ---------------------------------------------------------------------------
__global__ void pad_dist(const float* __restrict__ dist, float* __restrict__ DP) {
    const int total = BATCH * NNODE * NPAD;
    for (int idx = blockIdx.x * blockDim.x + threadIdx.x; idx < total;
         idx += gridDim.x * blockDim.x) {
        const int k   = idx % NPAD;
        const int rem = idx / NPAD;               // b*81 + m
        DP[idx] = (k < NNODE) ? dist[(long)rem * NNODE + k] : 0.0f;
    }
}

__global__ void pad_xg(const float* __restrict__ XG, float* __restrict__ XGP) {
    const int total = BATCH * NPAD * HIDC;
    for (int idx = blockIdx.x * blockDim.x + threadIdx.x; idx < total;
         idx += gridDim.x * blockDim.x) {
        const int h = idx % HIDC;
        const int r = (idx / HIDC) % NPAD;
        const int b = idx / (HIDC * NPAD);
        XGP[idx] = (r < NNODE) ? XG[((long)b * NNODE + r) * HIDC + h] : 0.0f;
    }
}

// ---------------------------------------------------------------------------
// BatchNorm statistics over a tensor viewed as (R, C, Ir), channel = middle.
// One block (256 threads) per channel.  stats[2c] = mean, stats[2c+1] = rstd.
// ---------------------------------------------------------------------------
__global__ void bn_stats(const float* __restrict__ Xv, float* __restrict__ stats,
                         int R, int C, int Ir) {
    const int c   = blockIdx.x;
    const int tid = threadIdx.x;
    const int total = R * Ir;
    float s = 0.0f, s2 = 0.0f;
    for (int j = tid; j < total; j += blockDim.x) {
        const int r = j / Ir;
        const int i = j - r * Ir;
        const float v = Xv[((long)r * C + c) * Ir + i];
        s += v;
        s2 += v * v;
    }
    __shared__ float sh[256];
    __shared__ float sh2[256];
    sh[tid] = s; sh2[tid] = s2;
    __syncthreads();
    for (int st = 128; st > 0; st >>= 1) {
        if (tid < st) { sh[tid] += sh[tid + st]; sh2[tid] += sh2[tid + st]; }
        __syncthreads();
    }
    if (tid == 0) {
        const float inv = 1.0f / (float)total;
        const float m   = sh[0] * inv;
        const float var = sh2[0] * inv - m * m;
        stats[2 * c]     = m;
        stats[2 * c + 1] = rsqrtf(var + EPSV);
    }
}

// ---------------------------------------------------------------------------
// Out = alpha * mish(g[c]*(X - mean[c])*rstd[c] + b[c])  (+ Out if accumulate)
// ---------------------------------------------------------------------------
__global__ void bn_apply_mish(const float* __restrict__ Xv,
                              const float* __restrict__ stats,
                              const float* __restrict__ g,
                              const float* __restrict__ bb,
                              float* __restrict__ Out,
                              int R, int C, int Ir, float alpha, int accumulate) {
    const int total = R * C * Ir;
    for (int idx = blockIdx.x * blockDim.x + threadIdx.x; idx < total;
         idx += gridDim.x * blockDim.x) {
        const int c = (idx / Ir) % C;
        const float v = Xv[idx];
        const float y = g[c] * (v - stats[2 * c]) * stats[2 * c + 1] + bb[c];
        const float m = alpha * mishf(y);
        Out[idx] = accumulate ? (Out[idx] + m) : m;
    }
}

// ---------------------------------------------------------------------------
// GCN per-node affine:  AGG = aw[n,0]*AGG + aw[n,1]*XG + ab[n]
// ---------------------------------------------------------------------------
__global__ void gcn_affine(float* __restrict__ AGG, const float* __restrict__ XG,
                           const float* __restrict__ aw, const float* __restrict__ ab,
                           int total) {
    for (int idx = blockIdx.x * blockDim.x + threadIdx.x; idx < total;
         idx += gridDim.x * blockDim.x) {
        const int n = (idx / HIDC) % NNODE;
        AGG[idx] = aw[2 * n] * AGG[idx] + aw[2 * n + 1] * XG[idx] + ab[n];
    }
}

// X = 0.5 * (XG + ACC)
__global__ void combine_half(float* __restrict__ X, const float* __restrict__ XG,
                             const float* __restrict__ ACC, int total) {
    for (int idx = blockIdx.x * blockDim.x + threadIdx.x; idx < total;
         idx += gridDim.x * blockDim.x) {
        X[idx] = 0.5f * (XG[idx] + ACC[idx]);
    }
}

// ---------------------------------------------------------------------------
// Deformable-conv bilinear gather -> im2col buffer V.
// Pixel coords: n = Xc*L + Yc ; conv grid value xc_in[b,c,y,x] = X[b, x*L+y, c].
// One thread per (pixel-in-chunk, k); float4 loop over 128 channels.
// V[row][c*K + k] matches W flat [o][c][k].
// ---------------------------------------------------------------------------
__global__ void deform_gather(const float* __restrict__ X,
                              const float* __restrict__ off,   // (B, 2K, L, L) slice
                              float* __restrict__ V,
                              int bBase, int K, int kwid, int pad) {
    const int p2 = blockIdx.x;                  // 0 .. CH_B*NNODE-1
    const int k  = threadIdx.x;
    if (k >= K) return;
    const int b  = bBase + p2 / NNODE;
    const int n  = p2 % NNODE;
    const int Xc = n / LDIM;
    const int Yc = n % LDIM;
    const int ky = k / kwid;
    const int kx = k % kwid;

    const float* offb = off + (long)b * (2 * K) * (LDIM * LDIM);
    const float oy = offb[(2 * k)     * (LDIM * LDIM) + Yc * LDIM + Xc];
    const float ox = offb[(2 * k + 1) * (LDIM * LDIM) + Yc * LDIM + Xc];

    const float py = (float)Yc + (float)ky - (float)pad + oy;
    const float px = (float)Xc + (float)kx - (float)pad + ox;
    const float y0f = floorf(py), x0f = floorf(px);
    const int   y0 = (int)y0f,    x0 = (int)x0f;
    const float wy1 = py - y0f,   wx1 = px - x0f;
    const float wy0 = 1.0f - wy1, wx0 = 1.0f - wx1;

    const int   ys[2]  = {y0, y0 + 1};
    const int   xs[2]  = {x0, x0 + 1};
    const float wys[2] = {wy0, wy1};
    const float wxs[2] = {wx0, wx1};

    float w[4];
    int   nidx[4];
    int q = 0;
    for (int qy = 0; qy < 2; ++qy) {
        for (int qx = 0; qx < 2; ++qx) {
            const int yi = ys[qy], xi = xs[qx];
            const bool valid = (yi >= 0) && (yi < LDIM) && (xi >= 0) && (xi < LDIM);
            const int yc2 = min(max(yi, 0), LDIM - 1);
            const int xc2 = min(max(xi, 0), LDIM - 1);
            w[q]    = valid ? wys[qy] * wxs[qx] : 0.0f;
            nidx[q] = xc2 * LDIM + yc2;           // node index of gathered pixel
            ++q;
        }
    }

    const float* xb = X + (long)b * NNODE * HIDC;
    const float4* g0 = (const float4*)(xb + (long)nidx[0] * HIDC);
    const float4* g1 = (const float4*)(xb + (long)nidx[1] * HIDC);
    const float4* g2 = (const float4*)(xb + (long)nidx[2] * HIDC);
    const float4* g3 = (const float4*)(xb + (long)nidx[3] * HIDC);
    float* vrow = V + (long)p2 * (HIDC * K) + k;
#pragma unroll 4
    for (int c4 = 0; c4 < HIDC / 4; ++c4) {
        const float4 a = g0[c4], bq = g1[c4], cq = g2[c4], dq = g3[c4];
        vrow[(long)(4 * c4 + 0) * K] = w[0] * a.x + w[1] * bq.x + w[2] * cq.x + w[3] * dq.x;
        vrow[(long)(4 * c4 + 1) * K] = w[0] * a.y + w[1] * bq.y + w[2] * cq.y + w[3] * dq.y;
        vrow[(long)(4 * c4 + 2) * K] = w[0] * a.z + w[1] * bq.z + w[2] * cq.z + w[3] * dq.z;
        vrow[(long)(4 * c4 + 3) * K] = w[0] * a.w + w[1] * bq.w + w[2] * cq.w + w[3] * dq.w;
    }
}

// ---------------------------------------------------------------------------
// Head kernels (tiny: 32x128 / 32x16 outputs)
// ---------------------------------------------------------------------------
__global__ void fpc_kernel(const float* __restrict__ X, const float* __restrict__ fw,
                           const float* __restrict__ fb, float* __restrict__ XC1) {
    const int b = blockIdx.x, c = threadIdx.x;
    float s = fb[c];
    for (int n = 0; n < NNODE; ++n)
        s += X[((long)b * NNODE + n) * HIDC + c] * fw[c * NNODE + n];
    XC1[b * HIDC + c] = s;
}

__global__ void fpg_kernel(const float* __restrict__ X, const float* __restrict__ dist,
                           const float* __restrict__ fw, const float* __restrict__ fbv,
                           float* __restrict__ XGF) {
    const int b = blockIdx.x, h = threadIdx.x;
    const int ctr = (NNODE - 1) / 2;   // 40
    const float* drow = dist + (long)b * NNODE * NNODE + (long)ctr * NNODE;
    float s = 0.0f;
    for (int m = 0; m < NNODE; ++m)
        s += drow[m] * X[((long)b * NNODE + m) * HIDC + h];
    const float res = X[((long)b * NNODE + ctr) * HIDC + h];
    XGF[b * HIDC + h] = fw[0] * s + fw[1] * res + fbv[0];
}

__global__ void head_out(const float* __restrict__ XC1, const float* __restrict__ XGF,
                         const float* __restrict__ ow, const float* __restrict__ ob,
                         float* __restrict__ out) {
    const int b = blockIdx.x, o = threadIdx.x;
    if (o >= NCLSO) return;
    float s = ob[o];
    for (int h = 0; h < HIDC; ++h)
        s += 0.5f * (XC1[b * HIDC + h] + XGF[b * HIDC + h]) * ow[o * HIDC + h];
    out[b * NCLSO + o] = s;
}

// ---------------------------------------------------------------------------
// Host orchestration
// ---------------------------------------------------------------------------
extern "C" void kernel_launch(void* const* d_in, const int* in_sizes, int n_in,
                              void* d_out, int out_size, void* d_ws, size_t ws_size,
                              hipStream_t stream) {
    (void)in_sizes; (void)n_in; (void)out_size; (void)ws_size;

    const float* x_in    = (const float*)d_in[0];   // (B, N, S)
    const float* dist    = (const float*)d_in[1];   // (B, N, N)
    const float* sp_w1   = (const float*)d_in[2];
    const float* sp_b1   = (const float*)d_in[3];
    const float* sp_g1   = (const float*)d_in[4];
    const float* sp_be1  = (const float*)d_in[5];
    const float* sp_w2   = (const float*)d_in[6];
    const float* sp_b2   = (const float*)d_in[7];
    const float* sp_g2   = (const float*)d_in[8];
    const float* sp_be2  = (const float*)d_in[9];
    const float* dcw_k3  = (const float*)d_in[10];
    const float* dcb_k3  = (const float*)d_in[11];
    const float* dcw_k5  = (const float*)d_in[12];
    const float* dcb_k5  = (const float*)d_in[13];
    const float* dcw_k7  = (const float*)d_in[14];
    const float* dcb_k7  = (const float*)d_in[15];
    const float* bnc_g   = (const float*)d_in[16];
    const float* bnc_b   = (const float*)d_in[17];
    const float* gcn_w   = (const float*)d_in[18];
    const float* gcn_b   = (const float*)d_in[19];
    const float* aff_w   = (const float*)d_in[20];
    const float* aff_b   = (const float*)d_in[21];
    const float* bng_g   = (const float*)d_in[22];
    const float* bng_b   = (const float*)d_in[23];
    const float* fpc_w   = (const float*)d_in[24];
    const float* fpc_b   = (const float*)d_in[25];
    const float* bnfpc_g = (const float*)d_in[26];
    const float* bnfpc_b = (const float*)d_in[27];
    const float* fpaff_w = (const float*)d_in[28];
    const float* fpaff_b = (const float*)d_in[29];
    const float* bnfpg_g = (const float*)d_in[30];
    const float* bnfpg_b = (const float*)d_in[31];
    const float* out_w   = (const float*)d_in[32];
    const float* out_b   = (const float*)d_in[33];
    const float* off_k3  = (const float*)d_in[34];  // (3, B, 18, 9, 9)
    const float* off_k5  = (const float*)d_in[35];  // (3, B, 50, 9, 9)
    const float* off_k7  = (const float*)d_in[36];  // (3, B, 98, 9, 9)

    // Workspace layout (floats): ~20 MB total
    float* Y1    = (float*)d_ws;                        // 2592*256
    float* Xb    = Y1    + (long)ROWS * 256;            // 2592*128 (current x)
    float* XG    = Xb    + (long)ROWS * HIDC;
    float* AGG   = XG    + (long)ROWS * HIDC;
    float* ACC   = AGG   + (long)ROWS * HIDC;
    float* YD    = ACC   + (long)ROWS * HIDC;
    float* DP    = YD    + (long)ROWS * HIDC;           // B*81*84 padded dist
    float* XGP   = DP    + (long)BATCH * NNODE * NPAD;  // B*84*128 padded xg
    float* V     = XGP   + (long)BATCH * NPAD * HIDC;   // CH_B*81*6272 max
    float* STATS = V     + (long)CH_B * NNODE * (HIDC * 49);
    float* XC1   = STATS + 2 * 256;
    float* XGF   = XC1   + BATCH * HIDC;

    const dim3 wblk(32, 4);   // 4 waves: 128 M-rows per block, 32 N-cols

    // Zero-padded copy of dist (K: 81 -> 84), used by all 3 iterations.
    pad_dist<<<512, 256, 0, stream>>>(dist, DP);

    // ---- spatial FC 1: (2592x200) @ (256x200)^T -> Y1, BN+mish ----
    wmma_gemm_f32<true><<<dim3((ROWS + 127) / 128, 256 / 32, 1), wblk, 0, stream>>>(
        x_in, sp_w1, Y1, sp_b1, ROWS, 256, SDIM, 0, 0, 0);
    bn_stats<<<256, 256, 0, stream>>>(Y1, STATS, ROWS, 256, 1);
    bn_apply_mish<<<1024, 256, 0, stream>>>(Y1, STATS, sp_g1, sp_be1, Y1,
                                            ROWS, 256, 1, 1.0f, 0);

    // ---- spatial FC 2: (2592x256) @ (128x256)^T -> Xb, BN+mish ----
    wmma_gemm_f32<true><<<dim3((ROWS + 127) / 128, HIDC / 32, 1), wblk, 0, stream>>>(
        Y1, sp_w2, Xb, sp_b2, ROWS, HIDC, 256, 0, 0, 0);
    bn_stats<<<HIDC, 256, 0, stream>>>(Xb, STATS, ROWS, HIDC, 1);
    bn_apply_mish<<<1024, 256, 0, stream>>>(Xb, STATS, sp_g2, sp_be2, Xb,
                                            ROWS, HIDC, 1, 1.0f, 0);

    struct KS { const float* w; const float* bi; const float* off;
                int K; int kw; int pad; long offStride; };
    const KS ks[3] = {
        { dcw_k3, dcb_k3, off_k3,  9, 3, 1, (long)BATCH * 18 * 81 },
        { dcw_k5, dcb_k5, off_k5, 25, 5, 2, (long)BATCH * 50 * 81 },
        { dcw_k7, dcb_k7, off_k7, 49, 7, 3, (long)BATCH * 98 * 81 },
    };

    for (int i = 0; i < 3; ++i) {
        // xg = x @ gcn_w[i]^T + gcn_b[i]
        wmma_gemm_f32<true><<<dim3((ROWS + 127) / 128, HIDC / 32, 1), wblk, 0, stream>>>(
            Xb, gcn_w + (long)i * HIDC * HIDC, XG, gcn_b + (long)i * HIDC,
            ROWS, HIDC, HIDC, 0, 0, 0);

        // agg[b] = dist[b] @ xg[b]  via zero-padded K=84 (branchless GEMM)
        pad_xg<<<512, 256, 0, stream>>>(XG, XGP);
        wmma_gemm_f32<false><<<dim3(1, HIDC / 32, BATCH), wblk, 0, stream>>>(
            DP, XGP, AGG, nullptr, NNODE, HIDC, NPAD,
            (long)NNODE * NPAD, (long)NPAD * HIDC, (long)NNODE * HIDC);

        // per-node affine, BN (per node over B*H) + mish -> XG
        gcn_affine<<<1024, 256, 0, stream>>>(AGG, XG,
            aff_w + (long)i * NNODE * 2, aff_b + (long)i * NNODE, ROWS * HIDC);
        bn_stats<<<NNODE, 256, 0, stream>>>(AGG, STATS, BATCH, NNODE, HIDC);
        bn_apply_mish<<<1024, 256, 0, stream>>>(AGG, STATS,
            bng_g + (long)i * NNODE, bng_b + (long)i * NNODE, XG,
            BATCH, NNODE, HIDC, 1.0f, 0);

        // deformable convs: gather -> im2col GEMM -> BN + mish, accumulate /3
        for (int j = 0; j < 3; ++j) {
            const int  K    = ks[j].K;
            const int  Kdim = HIDC * K;
            const float* offp = ks[j].off + (long)i * ks[j].offStride;
            const float* wp   = ks[j].w   + (long)i * HIDC * HIDC * K;
            const float* bp   = ks[j].bi  + (long)i * HIDC;
            for (int b0 = 0; b0 < BATCH; b0 += CH_B) {
                deform_gather<<<dim3(CH_B * NNODE), dim3(64), 0, stream>>>(
                    Xb, offp, V, b0, K, ks[j].kw, ks[j].pad);
                wmma_gemm_f32<true><<<dim3((CH_B * NNODE + 127) / 128, HIDC / 32, 1),
                                      wblk, 0, stream>>>(
                    V, wp, YD + (long)b0 * NNODE * HIDC, bp,
                    CH_B * NNODE, HIDC, Kdim, 0, 0, 0);
            }
            bn_stats<<<HIDC, 256, 0, stream>>>(YD, STATS, ROWS, HIDC, 1);
            bn_apply_mish<<<1024, 256, 0, stream>>>(YD, STATS,
                bnc_g + (long)(i * 3 + j) * HIDC, bnc_b + (long)(i * 3 + j) * HIDC,
                ACC, ROWS, HIDC, 1, 1.0f / 3.0f, (j > 0) ? 1 : 0);
        }

        // x = (xg + xc) / 2
        combine_half<<<1024, 256, 0, stream>>>(Xb, XG, ACC, ROWS * HIDC);
    }

    // ---- head ----
    fpc_kernel<<<BATCH, HIDC, 0, stream>>>(Xb, fpc_w, fpc_b, XC1);
    bn_stats<<<HIDC, 256, 0, stream>>>(XC1, STATS, BATCH, HIDC, 1);
    bn_apply_mish<<<64, 256, 0, stream>>>(XC1, STATS, bnfpc_g, bnfpc_b, XC1,
                                          BATCH, HIDC, 1, 1.0f, 0);

    fpg_kernel<<<BATCH, HIDC, 0, stream>>>(Xb, dist, fpaff_w, fpaff_b, XGF);
    bn_stats<<<1, 256, 0, stream>>>(XGF, STATS, BATCH, 1, HIDC);
    bn_apply_mish<<<64, 256, 0, stream>>>(XGF, STATS, bnfpg_g, bnfpg_b, XGF,
                                          BATCH, 1, HIDC, 1.0f, 0);

    head_out<<<BATCH, 32, 0, stream>>>(XC1, XGF, out_w, out_b, (float*)d_out);
}